// TorchComputeGradientModule_59605556133982
// MI455X (gfx1250) — compile-verified
//
#include <hip/hip_runtime.h>

// MI455X / gfx1250, wave32. One wave per batch element.
// Memory-bound kernel (~532 B/elem -> ~23us roofline @ 23.3 TB/s).
// 9x9 matvec y = W @ F_normalised done via 3x V_WMMA_F32_16X16X4_F32
// (A = F broadcast across rows, B = W^T chunk, all D rows == y).

typedef float v2f __attribute__((ext_vector_type(2)));
typedef float v8f __attribute__((ext_vector_type(8)));

#define WPB 8  // waves per 256-thread block

__global__ __launch_bounds__(256) void epi_fgrad_wmma(
    const float* __restrict__ gR1, const float* __restrict__ gR2,
    const float* __restrict__ gt1, const float* __restrict__ gt2,
    const float* __restrict__ gf1, const float* __restrict__ gf2,
    const float* __restrict__ gW,
    float* __restrict__ out, float* __restrict__ partial, long long Bn)
{
    __shared__ float wsum[WPB];
    const int  lane = threadIdx.x & 31;
    const int  wv   = threadIdx.x >> 5;
    const long long b = (long long)blockIdx.x * WPB + wv;   // uniform per wave

    float halfloss = 0.0f;

    if (b < Bn) {
        // ---------- per-element small inputs (wave-uniform addresses) ----------
        float R1m[3][3], R2m[3][3];
        #pragma unroll
        for (int i = 0; i < 3; ++i) {
            #pragma unroll
            for (int j = 0; j < 3; ++j) {
                R1m[i][j] = gR1[b * 9 + i * 3 + j];
                R2m[i][j] = gR2[b * 9 + i * 3 + j];
            }
        }
        float t1v[3], t2v[3];
        #pragma unroll
        for (int i = 0; i < 3; ++i) { t1v[i] = gt1[b * 3 + i]; t2v[i] = gt2[b * 3 + i]; }
        const float f1 = gf1[b];
        const float f2 = gf2[b];

        // ---------- forward: R_rel = R2 * R1^T ----------
        float Rr[3][3];
        #pragma unroll
        for (int i = 0; i < 3; ++i)
            #pragma unroll
            for (int j = 0; j < 3; ++j)
                Rr[i][j] = R2m[i][0]*R1m[j][0] + R2m[i][1]*R1m[j][1] + R2m[i][2]*R1m[j][2];

        const float T1[3][3] = {{0.f,-t1v[2],t1v[1]},{t1v[2],0.f,-t1v[0]},{-t1v[1],t1v[0],0.f}};
        const float T2[3][3] = {{0.f,-t2v[2],t2v[1]},{t2v[2],0.f,-t2v[0]},{-t2v[1],t2v[0],0.f}};

        // essential E = R_rel*T1 - T2*R_rel
        float Em[3][3];
        #pragma unroll
        for (int i = 0; i < 3; ++i)
            #pragma unroll
            for (int j = 0; j < 3; ++j)
                Em[i][j] = (Rr[i][0]*T1[0][j] + Rr[i][1]*T1[1][j] + Rr[i][2]*T1[2][j])
                         - (T2[i][0]*Rr[0][j] + T2[i][1]*Rr[1][j] + T2[i][2]*Rr[2][j]);

        const float K1v[3] = {f1, f1, 1.0f};
        const float K2v[3] = {f2, f2, 1.0f};

        float Ff[9];
        #pragma unroll
        for (int i = 0; i < 3; ++i)
            #pragma unroll
            for (int j = 0; j < 3; ++j)
                Ff[i*3 + j] = K2v[i] * Em[i][j] * K1v[j];

        float ss = 0.f;
        #pragma unroll
        for (int k = 0; k < 9; ++k) ss += Ff[k] * Ff[k];
        const float nrm  = sqrtf(ss) + 1e-8f;
        const float rinv = 1.0f / nrm;

        float Fn[12];                       // padded to K=12 so all indices stay in-bounds
        #pragma unroll
        for (int k = 0; k < 9; ++k) Fn[k] = Ff[k] * rinv;
        Fn[9] = Fn[10] = Fn[11] = 0.0f;

        // ---------- WMMA: y = W @ Fn via 3 x 16x16x4 f32 ----------
        // A layout (ISA 7.12.2, 32-bit A 16x4): lane half selects K{0,1} vs K{2,3},
        // VGPR index iterates K inside the half. A is F broadcast over rows M.
        // B layout (mirrors A / 16-bit B pattern): N = lane&15,
        // K = vgpr + 2*(lane>=16). B holds W^T (rows/cols >= 9 zero-masked).
        const bool hi  = lane >= 16;
        const int  n16 = lane & 15;
        const bool nok = n16 < 9;
        const int  nc  = nok ? n16 : 8;                 // clamped -> always in-bounds
        const float* Wb = gW + b * 81;

        v8f acc = {0.f, 0.f, 0.f, 0.f, 0.f, 0.f, 0.f, 0.f};
        #pragma unroll
        for (int c = 0; c < 3; ++c) {
            v2f Av;
            Av.x = hi ? Fn[4*c + 2] : Fn[4*c + 0];
            Av.y = hi ? Fn[4*c + 3] : Fn[4*c + 1];

            const int kb  = 4*c + (hi ? 2 : 0);
            const int k0c = (kb     < 9) ? kb     : 8;
            const int k1c = (kb + 1 < 9) ? kb + 1 : 8;
            float w0 = __builtin_nontemporal_load(Wb + nc * 9 + k0c);
            float w1 = __builtin_nontemporal_load(Wb + nc * 9 + k1c);
            w0 = (nok && (kb     < 9)) ? w0 : 0.0f;
            w1 = (nok && (kb + 1 < 9)) ? w1 : 0.0f;
            v2f Bv; Bv.x = w0; Bv.y = w1;

            // 8 args: (neg_a, A, neg_b, B, c_mod, C, reuse_a, reuse_b)
            acc = __builtin_amdgcn_wmma_f32_16x16x4_f32(
                      false, Av, false, Bv, (short)0, acc, false, false);
        }

        // every D row equals y -> acc[0] in lane l holds y[l & 15]
        const float yl = acc[0];
        float y[9];
        #pragma unroll
        for (int j = 0; j < 9; ++j) y[j] = __shfl(yl, j, 32);

        float s = 0.f;
        #pragma unroll
        for (int j = 0; j < 9; ++j) s += Fn[j] * y[j];
        halfloss = 0.5f * s;

        // ---------- backward ----------
        float dFf[9];
        #pragma unroll
        for (int k = 0; k < 9; ++k) dFf[k] = (y[k] - s * Fn[k]) * rinv;

        float dE[3][3];
        float df1 = 0.f, df2 = 0.f;
        #pragma unroll
        for (int i = 0; i < 3; ++i) {
            #pragma unroll
            for (int j = 0; j < 3; ++j) {
                const float d  = dFf[i*3 + j];
                dE[i][j]       = d * K2v[i] * K1v[j];
                const float de = d * Em[i][j];
                if (j < 2) df1 += de * K2v[i];   // sum d_K1_inv[:, :, :2]
                if (i < 2) df2 += de * K1v[j];   // sum d_K2_inv[:, :2, :]
            }
        }

        // d_R_rel = dE*T1^T - T2^T*dE ; d_t1_x = Rr^T*dE ; d_t2_x = -dE*Rr^T
        float dRr[3][3], dT1x[3][3], dT2x[3][3];
        #pragma unroll
        for (int i = 0; i < 3; ++i) {
            #pragma unroll
            for (int j = 0; j < 3; ++j) {
                dRr[i][j]  = (dE[i][0]*T1[j][0] + dE[i][1]*T1[j][1] + dE[i][2]*T1[j][2])
                           - (T2[0][i]*dE[0][j] + T2[1][i]*dE[1][j] + T2[2][i]*dE[2][j]);
                dT1x[i][j] =  Rr[0][i]*dE[0][j] + Rr[1][i]*dE[1][j] + Rr[2][i]*dE[2][j];
                dT2x[i][j] = -(dE[i][0]*Rr[j][0] + dE[i][1]*Rr[j][1] + dE[i][2]*Rr[j][2]);
            }
        }
        const float dt1v[3] = {dT1x[2][1]-dT1x[1][2], dT1x[0][2]-dT1x[2][0], dT1x[1][0]-dT1x[0][1]};
        const float dt2v[3] = {dT2x[2][1]-dT2x[1][2], dT2x[0][2]-dT2x[2][0], dT2x[1][0]-dT2x[0][1]};

        // d_R1 = dRr^T * R2 ; d_R2 = dRr * R1
        float g1[9], g2[9];
        #pragma unroll
        for (int i = 0; i < 3; ++i) {
            #pragma unroll
            for (int j = 0; j < 3; ++j) {
                g1[i*3+j] = dRr[0][i]*R2m[0][j] + dRr[1][i]*R2m[1][j] + dRr[2][i]*R2m[2][j];
                g2[i*3+j] = dRr[i][0]*R1m[0][j] + dRr[i][1]*R1m[1][j] + dRr[i][2]*R1m[2][j];
            }
        }

        // ---------- scatter the 26 outputs: lane l writes output l ----------
        const float o =
            lane==0  ? g1[0] : lane==1  ? g1[1] : lane==2  ? g1[2] :
            lane==3  ? g1[3] : lane==4  ? g1[4] : lane==5  ? g1[5] :
            lane==6  ? g1[6] : lane==7  ? g1[7] : lane==8  ? g1[8] :
            lane==9  ? g2[0] : lane==10 ? g2[1] : lane==11 ? g2[2] :
            lane==12 ? g2[3] : lane==13 ? g2[4] : lane==14 ? g2[5] :
            lane==15 ? g2[6] : lane==16 ? g2[7] : lane==17 ? g2[8] :
            lane==18 ? dt1v[0] : lane==19 ? dt1v[1] : lane==20 ? dt1v[2] :
            lane==21 ? dt2v[0] : lane==22 ? dt2v[1] : lane==23 ? dt2v[2] :
            lane==24 ? df1 : df2;

        const unsigned long long BB = (unsigned long long)Bn;
        const unsigned long long bb = (unsigned long long)b;
        unsigned long long off;
        if      (lane <  9) off = 1ull +            9ull*bb + (unsigned)lane;
        else if (lane < 18) off = 1ull +  9ull*BB + 9ull*bb + (unsigned)(lane - 9);
        else if (lane < 21) off = 1ull + 18ull*BB + 3ull*bb + (unsigned)(lane - 18);
        else if (lane < 24) off = 1ull + 21ull*BB + 3ull*bb + (unsigned)(lane - 21);
        else if (lane == 24) off = 1ull + 24ull*BB + bb;
        else                 off = 1ull + 25ull*BB + bb;
        if (lane < 26) __builtin_nontemporal_store(o, out + off);
    }

    // ---------- deterministic loss reduction (no float atomics) ----------
    if (lane == 0) wsum[wv] = halfloss;
    __syncthreads();
    if (threadIdx.x == 0) {
        float acc = 0.f;
        #pragma unroll
        for (int w = 0; w < WPB; ++w) acc += wsum[w];
        partial[blockIdx.x] = acc;
    }
}

__global__ __launch_bounds__(256) void reduce_partials(
    const float* __restrict__ partial, int n, float* __restrict__ out)
{
    __shared__ float sm[256];
    float s = 0.f;
    for (int i = threadIdx.x; i < n; i += 256) s += partial[i];
    sm[threadIdx.x] = s;
    __syncthreads();
    #pragma unroll
    for (int st = 128; st > 0; st >>= 1) {
        if ((int)threadIdx.x < st) sm[threadIdx.x] += sm[threadIdx.x + st];
        __syncthreads();
    }
    if (threadIdx.x == 0) out[0] = sm[0];
}

extern "C" void kernel_launch(void* const* d_in, const int* in_sizes, int n_in,
                              void* d_out, int out_size, void* d_ws, size_t ws_size,
                              hipStream_t stream)
{
    (void)n_in; (void)out_size; (void)ws_size;
    const float* R1 = (const float*)d_in[0];
    const float* R2 = (const float*)d_in[1];
    const float* t1 = (const float*)d_in[2];
    const float* t2 = (const float*)d_in[3];
    const float* f1 = (const float*)d_in[4];
    const float* f2 = (const float*)d_in[5];
    const float* W  = (const float*)d_in[6];
    const long long Bn = (long long)in_sizes[4];          // f1_inv has B elements
    const int nblk = (int)((Bn + WPB - 1) / WPB);
    float* partial = (float*)d_ws;                        // nblk floats of scratch

    epi_fgrad_wmma<<<nblk, 256, 0, stream>>>(R1, R2, t1, t2, f1, f2, W,
                                             (float*)d_out, partial, Bn);
    reduce_partials<<<1, 256, 0, stream>>>(partial, nblk, (float*)d_out);
}